// SimplePACEdgeMode_42125039239772
// MI455X (gfx1250) — compile-verified
//
#include <hip/hip_runtime.h>
#include <math.h>

#ifndef M_PI
#define M_PI 3.14159265358979323846
#endif

typedef float v2f __attribute__((ext_vector_type(2)));
typedef float v8f __attribute__((ext_vector_type(8)));
typedef unsigned int u32x4 __attribute__((ext_vector_type(4)));
typedef int          i32x4 __attribute__((ext_vector_type(4)));
typedef int          i32x8 __attribute__((ext_vector_type(8)));

#define NSIG   128      // B*C*S = 4*8*4
#define TLEN   2048
#define NB     20
#define LTAP   769
#define KPAD   772      // taps padded to multiple of 4 for 16x16x4 k-loop
#define NBPAD  32       // bands padded to 2 N-tiles of 16
#define REFL   768      // reflect pad
#define PADLEN 3584     // TLEN + 2*REFL
#define NBINS  18
#define MROWS  64       // t-rows per block (4 WMMA tiles per wave)
#define PATCHN 840      // staged floats: (MROWS-16) + 788, padded

#if __has_builtin(__builtin_amdgcn_tensor_load_to_lds)
#define HAVE_TDM 1
#endif

// ---------------------------------------------------------------------------
// 1) Symmetrized, padded filter bank: fwd+bwd zero-phase filtering collapses
//    to a single correlation with 0.5*(K[k]+K[L-1-k]).
// ---------------------------------------------------------------------------
__global__ void build_ksym(const float* __restrict__ K, float* __restrict__ ksym) {
    int i = blockIdx.x * blockDim.x + threadIdx.x;
    if (i >= NBPAD * KPAD) return;
    int band = i / KPAD, tap = i % KPAD;
    float v = 0.f;
    if (band < NB && tap < LTAP)
        v = 0.5f * (K[band * LTAP + tap] + K[band * LTAP + (LTAP - 1 - tap)]);
    ksym[i] = v;
}

// ---------------------------------------------------------------------------
// 2) Reflect-pad signals (mode='reflect', edge not repeated).
// ---------------------------------------------------------------------------
__global__ void build_pad(const float* __restrict__ x, float* __restrict__ P) {
    int i = blockIdx.x * blockDim.x + threadIdx.x;
    if (i >= NSIG * PADLEN) return;
    int s = i / PADLEN, j = i % PADLEN;
    int t = j - REFL;
    if (t < 0) t = -t;
    if (t > TLEN - 1) t = 2 * (TLEN - 1) - t;
    P[i] = x[s * TLEN + t];
}

// ---------------------------------------------------------------------------
// 3) FIR bank as implicit GEMM on V_WMMA_F32_16X16X4_F32.
//    filt[s][b][t] = sum_k Ksym[b][k] * P[s][t+384+k]
//    Block = one signal x 64 t-rows; 2 waves cover 32 (padded) bands; each
//    wave runs 4 WMMA tiles per k-step so one B fragment feeds 4 matrix ops.
//    Patch staged to LDS via the Tensor Data Mover (1-D D# descriptor) when
//    the builtin is available, else plain loads.
//    A 16x4 layout: lanes 0-15 -> M rows, K={0,1}; lanes 16-31 -> K={2,3}.
//    B 4x16 mirrors A with N=lane%16. D: VGPR j -> M=j(+8), N=lane%16.
// ---------------------------------------------------------------------------
__global__ void __launch_bounds__(64)
conv_wmma(const float* __restrict__ P, const float* __restrict__ ksym,
          float* __restrict__ filt) {
    __shared__ float patch[PATCHN];        // P[s][t0+384 .. t0+384+835]
    const int t0   = blockIdx.x * MROWS;
    const int s    = blockIdx.y;
    const int tid  = threadIdx.x;
    const int lane = tid & 31;
    const int wave = tid >> 5;

    const float* gsrc = P + (size_t)s * PADLEN + t0 + 384;

#if defined(HAVE_TDM)
    if (wave == 0) {
        // ---- Tensor DMA descriptor: 1 row of 836 fp32 elements -> LDS ----
        unsigned long long ga = (unsigned long long)(uintptr_t)gsrc;
        unsigned lds_off = (unsigned)(unsigned long long)(uintptr_t)&patch[0];
        u32x4 g0;
        g0.x = 1u;                                     // count=1, user mode
        g0.y = lds_off;                                // lds_addr (bytes)
        g0.z = (unsigned)(ga & 0xFFFFFFFFull);         // global_addr[31:0]
        g0.w = (unsigned)((ga >> 32) & 0x1FFFFFFull)   // global_addr[56:32]
             | (2u << 30);                             // type = 2 ("image")
        i32x8 g1;
        g1[0] = (int)(2u << 16);                       // data_size=4B, mask=0
        g1[1] = (int)((836u & 0xFFFFu) << 16);         // tensor_dim0 lo16
        g1[2] = (int)(1u << 16);                       // dim0 hi=0 | tensor_dim1=1
        g1[3] = (int)(836u << 16);                     // dim1 hi=0 | tile_dim0=836
        g1[4] = 1;                                     // tile_dim1=1, tile_dim2=0
        g1[5] = 836;                                   // tensor_dim0_stride lo
        g1[6] = 0;                                     // stride hi | dim1_stride lo
        g1[7] = 0;
#if __has_include(<hip/amd_detail/amd_gfx1250_TDM.h>)
        i32x4 z4 = {}; i32x8 z8 = {};
        __builtin_amdgcn_tensor_load_to_lds(g0, g1, z4, z4, z8, 0);
#else
        i32x4 z4 = {};
        __builtin_amdgcn_tensor_load_to_lds(g0, g1, z4, z4, 0);
#endif
        __builtin_amdgcn_s_wait_tensorcnt(0);
    }
    __syncthreads();
#else
    for (int i = tid; i < 836; i += 64) patch[i] = gsrc[i];
    __syncthreads();
#endif

    const int n0   = wave * 16;
    const int hi   = (lane < 16) ? 0 : 1;  // lane half
    const int m    = lane & 15;            // A row (M) / B,D col (N)
    const int koff = hi * 2;               // K sub-offset within 4-tap step

    const float* kb = ksym + (size_t)(n0 + m) * KPAD;
    for (int pf = 0; pf < KPAD; pf += 128)
        __builtin_prefetch(kb + pf, 0, 0); // global_prefetch_b8 warms L2/WGP$

    v8f acc0 = {}, acc1 = {}, acc2 = {}, acc3 = {};
    for (int k = 0; k < KPAD; k += 4) {
        v2f b;
        b.x = kb[k + koff];
        b.y = kb[k + koff + 1];
        const int base = m + k + koff;
        v2f a0, a1, a2, a3;
        a0.x = patch[base];      a0.y = patch[base + 1];
        a1.x = patch[base + 16]; a1.y = patch[base + 17];
        a2.x = patch[base + 32]; a2.y = patch[base + 33];
        a3.x = patch[base + 48]; a3.y = patch[base + 49];
        acc0 = __builtin_amdgcn_wmma_f32_16x16x4_f32(false, a0, false, b, (short)0, acc0, false, false);
        acc1 = __builtin_amdgcn_wmma_f32_16x16x4_f32(false, a1, false, b, (short)0, acc1, false, false);
        acc2 = __builtin_amdgcn_wmma_f32_16x16x4_f32(false, a2, false, b, (short)0, acc2, false, false);
        acc3 = __builtin_amdgcn_wmma_f32_16x16x4_f32(false, a3, false, b, (short)0, acc3, false, false);
    }

    const int band = n0 + m;
    if (band < NB) {
        float* o = filt + ((size_t)s * NB + band) * TLEN + t0 + hi * 8;
#pragma unroll
        for (int j = 0; j < 8; ++j) o[j]      = acc0[j];
#pragma unroll
        for (int j = 0; j < 8; ++j) o[16 + j] = acc1[j];
#pragma unroll
        for (int j = 0; j < 8; ++j) o[32 + j] = acc2[j];
#pragma unroll
        for (int j = 0; j < 8; ++j) o[48 + j] = acc3[j];
    }
}

// ---------------------------------------------------------------------------
// 4) 2048-pt Hilbert analytic signal per (signal, band) in LDS.
// ---------------------------------------------------------------------------
__device__ __forceinline__ int brev11(int t) {
    return (int)(__brev((unsigned)t) >> 21);
}

__device__ __forceinline__ void fft2048_dit(float2* d, int tid) {
    for (int st = 0; st < 11; ++st) {
        const int hlen = 1 << st;
        for (int b = tid; b < TLEN / 2; b += 256) {
            int j  = b & (hlen - 1);
            int i0 = ((b >> st) << (st + 1)) + j;
            int i1 = i0 + hlen;
            float ang = -(float)M_PI * (float)j / (float)hlen;
            float c = __cosf(ang), sn = __sinf(ang);
            float2 u = d[i0], v = d[i1];
            float2 w = make_float2(v.x * c - v.y * sn, v.x * sn + v.y * c);
            d[i0] = make_float2(u.x + w.x, u.y + w.y);
            d[i1] = make_float2(u.x - w.x, u.y - w.y);
        }
        __syncthreads();
    }
}

__global__ void __launch_bounds__(256)
hilbert_fft(float* __restrict__ filt) {
    __shared__ float2 d[TLEN];             // 16 KB of the 320 KB WGP LDS
    const int band = blockIdx.x;           // 0..19
    const int s    = blockIdx.y;           // 0..127
    const int tid  = threadIdx.x;
    float* base = filt + ((size_t)s * NB + band) * TLEN;

    for (int t = tid; t < TLEN; t += 256)
        d[brev11(t)] = make_float2(base[t], 0.f);
    __syncthreads();

    fft2048_dit(d, tid);                   // forward FFT (natural order out)

    // analytic multiplier, then conj (inverse via conj-fft-conj)
    for (int t = tid; t < TLEN; t += 256) {
        float2 v = d[t];
        float g = (t == 0 || t == TLEN / 2) ? 1.f : (t < TLEN / 2 ? 2.f : 0.f);
        d[t] = make_float2(v.x * g, -v.y * g);
    }
    __syncthreads();

    // in-place bit-reverse permute (each unordered pair owned by one thread)
    for (int t = tid; t < TLEN; t += 256) {
        int r = brev11(t);
        if (t < r) { float2 tmp = d[t]; d[t] = d[r]; d[r] = tmp; }
    }
    __syncthreads();

    fft2048_dit(d, tid);                   // fft(conj(X))

    const float inv = 1.f / (float)TLEN;
    for (int t = tid; t < TLEN; t += 256) {
        float re =  d[t].x * inv;
        float im = -d[t].y * inv;          // conj + 1/N => ifft
        base[t] = (band < 10) ? atan2f(im, re) : sqrtf(re * re + im * im);
    }
}

// ---------------------------------------------------------------------------
// 5) Modulation index per signal: 18-bin phase histogram of amplitudes.
// ---------------------------------------------------------------------------
__global__ void __launch_bounds__(256)
mi_kernel(const float* __restrict__ filt, float* __restrict__ miw) {
    __shared__ float sums[10 * 10 * NBINS]; // [p][a][bin]
    __shared__ float cnts[10 * NBINS];      // [p][bin]
    const int s   = blockIdx.x;
    const int tid = threadIdx.x;
    for (int i = tid; i < 10 * 10 * NBINS; i += 256) sums[i] = 0.f;
    for (int i = tid; i < 10 * NBINS;      i += 256) cnts[i] = 0.f;
    __syncthreads();

    const float* base = filt + (size_t)s * NB * TLEN;
    const float binscale = (float)NBINS / (2.f * (float)M_PI);
    for (int t = tid; t < TLEN; t += 256) {
        float av[10];
#pragma unroll
        for (int a = 0; a < 10; ++a) av[a] = base[(10 + a) * TLEN + t];
#pragma unroll
        for (int p = 0; p < 10; ++p) {
            float ph  = base[p * TLEN + t];
            int bin = (int)floorf((ph + (float)M_PI) * binscale);
            bin = bin < 0 ? 0 : (bin > NBINS - 1 ? NBINS - 1 : bin);
            atomicAdd(&cnts[p * NBINS + bin], 1.f);
#pragma unroll
            for (int a = 0; a < 10; ++a)
                atomicAdd(&sums[(p * 10 + a) * NBINS + bin], av[a]);
        }
    }
    __syncthreads();

    if (tid < 100) {
        const int p = tid / 10, a = tid % 10;
        float mean[NBINS];
        float tot = 0.f;
        for (int b = 0; b < NBINS; ++b) {
            float c = cnts[p * NBINS + b];
            float m = sums[(p * 10 + a) * NBINS + b] / fmaxf(c, 1.f);
            mean[b] = m; tot += m;
        }
        tot = fmaxf(tot, 1e-12f);
        float ent = 0.f;
        for (int b = 0; b < NBINS; ++b) {
            float pn = mean[b] / tot;
            ent += pn * logf(fmaxf(pn, 1e-12f));
        }
        const float logn = logf((float)NBINS);
        miw[(size_t)s * 100 + tid] = (logn + ent) / logn;
    }
}

// ---------------------------------------------------------------------------
// 6) Mean over S (S is innermost in bcs ordering).
// ---------------------------------------------------------------------------
__global__ void finalize(const float* __restrict__ miw, float* __restrict__ out) {
    int i = blockIdx.x * blockDim.x + threadIdx.x;
    if (i >= 3200) return;
    int bc = i / 100, pa = i % 100;
    float acc = 0.f;
#pragma unroll
    for (int s = 0; s < 4; ++s) acc += miw[(size_t)(bc * 4 + s) * 100 + pa];
    out[i] = 0.25f * acc;
}

// ---------------------------------------------------------------------------
extern "C" void kernel_launch(void* const* d_in, const int* in_sizes, int n_in,
                              void* d_out, int out_size, void* d_ws, size_t ws_size,
                              hipStream_t stream) {
    const float* x = (const float*)d_in[0];   // 128*2048
    const float* K = (const float*)d_in[1];   // 20*769
    float* out = (float*)d_out;               // 3200

    // workspace layout (floats): ksym | padded signals | filt (reused as pha/amp) | mi
    float* ksym = (float*)d_ws;                               // 32*772      = 24704
    float* P    = ksym + NBPAD * KPAD;                        // 128*3584    = 458752
    float* filt = P + (size_t)NSIG * PADLEN;                  // 128*20*2048 = 5242880
    float* miw  = filt + (size_t)NSIG * NB * TLEN;            // 128*100     = 12800
    (void)in_sizes; (void)n_in; (void)out_size; (void)ws_size;

    build_ksym<<<(NBPAD * KPAD + 255) / 256, 256, 0, stream>>>(K, ksym);
    build_pad<<<(NSIG * PADLEN + 255) / 256, 256, 0, stream>>>(x, P);
    conv_wmma<<<dim3(TLEN / MROWS, NSIG), 64, 0, stream>>>(P, ksym, filt);
    hilbert_fft<<<dim3(NB, NSIG), 256, 0, stream>>>(filt);
    mi_kernel<<<NSIG, 256, 0, stream>>>(filt, miw);
    finalize<<<(3200 + 255) / 256, 256, 0, stream>>>(miw, out);
}